// RaycastRGBD_43104291782956
// MI455X (gfx1250) — compile-verified
//
#include <hip/hip_runtime.h>
#include <cstdint>

// Problem constants (match reference)
#define RW 320
#define RH 240
#define RD 96
#define N_CELLS (RD * RD * RD)   // 884736 (B=1)
#define N_PIX   (RW * RH)        // 76800
#define NSTEPS  191
#define DEPTH_MIN 0.5f
#define INC       0.5f
#define THRESH    1.0f
#define TRUNCV    100.0f

typedef __attribute__((ext_vector_type(2))) float v2f;
typedef __attribute__((ext_vector_type(8))) float v8f;

// ---------------------------------------------------------------------------
// 1) Init dense grid: each cell = { sdf = TRUNC (low 32b), idx = -1 (high 32b) }
// ---------------------------------------------------------------------------
__global__ __launch_bounds__(256) void init_grid_kernel(uint64_t* __restrict__ grid) {
  int i = blockIdx.x * 256 + threadIdx.x;
  if (i < N_CELLS) {
    uint64_t v = (0xFFFFFFFFull << 32) | (uint64_t)__float_as_uint(TRUNCV);
    grid[i] = v;
  }
}

// ---------------------------------------------------------------------------
// 2) Scatter sparse locs into the dense grid. Single b64 store keeps the
//    (sdf, idx) pair consistent if duplicate linear indices race.
// ---------------------------------------------------------------------------
__global__ __launch_bounds__(256) void scatter_kernel(const int* __restrict__ locs,
                                                      const float* __restrict__ vsdf,
                                                      uint64_t* __restrict__ grid,
                                                      int n) {
  int i = blockIdx.x * 256 + threadIdx.x;
  if (i >= n) return;
  int x = locs[i * 4 + 0];
  int y = locs[i * 4 + 1];
  int z = locs[i * 4 + 2];
  int b = locs[i * 4 + 3];
  int lin = (((b * RD + x) * RD + y) * RD + z);
  uint64_t v = ((uint64_t)(uint32_t)i << 32) | (uint64_t)__float_as_uint(vsdf[i]);
  grid[lin] = v;
}

// ---------------------------------------------------------------------------
// 3) Batched camera->world ray rotation via V_WMMA_F32_16X16X4_F32.
//    Each wave computes 16 rays: D(16x16) = A(16x4) * B(4x16),
//    A rows = normalized dir_cam (x,y,z,0), B = R^T padded (B[k][n] = R[n][k]).
//    ISA layout: A: lanes 0-15 M=0..15 K={0,1}, lanes 16-31 K={2,3}.
//                D: VGPR j, lanes 0-15 -> D[j][lane]; lanes 16-31 -> D[j+8][lane-16].
// ---------------------------------------------------------------------------
__global__ __launch_bounds__(256) void dir_kernel(const float* __restrict__ vm,
                                                  const float* __restrict__ intr,
                                                  float* __restrict__ dirs /* float4 per ray */) {
  int gtid = blockIdx.x * 256 + threadIdx.x;
  int wave = gtid >> 5;
  int lane = gtid & 31;
  int m    = lane & 15;
  bool hi  = (lane >= 16);
  int ray  = wave * 16 + m;

  float fx = intr[0], fy = intr[1], mx = intr[2], my = intr[3];
  int u = ray % RW;
  int v = ray / RW;

  // A operand: this lane's two K-slots of dir_cam for ray m
  float a0, a1;
  if (!hi) { a0 = ((float)u - mx) / fx; a1 = ((float)v - my) / fy; }  // K=0,1 (x,y)
  else     { a0 = 1.0f;                 a1 = 0.0f; }                   // K=2,3 (z,pad)

  // normalize dir_cam: lane pair (L, L^16) holds the 3 components of one ray
  float ss = fmaf(a0, a0, a1 * a1);
  ss += __shfl_xor(ss, 16, 32);          // x^2 + y^2 + 1
  float inv = 1.0f / sqrtf(ss);
  a0 *= inv;
  a1 *= inv;

  // B operand: B[k][n] = R[n][k] for n<3,k<3; else 0.  R[i][j] = vm[i*4+j].
  float b0 = 0.0f, b1 = 0.0f;
  if (m < 3) {
    if (!hi) { b0 = vm[m * 4 + 0]; b1 = vm[m * 4 + 1]; }  // K=0,1
    else     { b0 = vm[m * 4 + 2]; }                       // K=2 (K=3 pad)
  }

  v2f A;  A[0] = a0; A[1] = a1;
  v2f Bv; Bv[0] = b0; Bv[1] = b1;
  v8f C = {};
  v8f Dm = __builtin_amdgcn_wmma_f32_16x16x4_f32(
      /*neg_a=*/false, A, /*neg_b=*/false, Bv,
      /*c_mod=*/(short)0, C, /*reuse_a=*/false, /*reuse_b=*/false);

  // Columns 0..2 of D are dir_w components; lane (m<3) holds component m for
  // rays M = j + (hi?8:0) across the 8 result VGPRs.
  if (m < 3) {
    int baseRay = wave * 16 + (hi ? 8 : 0);
#pragma unroll
    for (int j = 0; j < 8; ++j) {
      dirs[(baseRay + j) * 4 + m] = Dm[j];
    }
  }
}

// ---------------------------------------------------------------------------
// 4) Ray march. One thread per pixel; wave32 = 8x4 pixel tile for L2/WGP$
//    locality. 4-wide unroll => 4 independent global_load_b64 in flight.
//    Wave-coherent early exit once all 32 lanes have their first crossing.
// ---------------------------------------------------------------------------
__global__ __launch_bounds__(256) void raycast_kernel(const uint64_t* __restrict__ grid,
                                                      const float4* __restrict__ dirs,
                                                      const float* __restrict__ vm,
                                                      const float* __restrict__ vcol,
                                                      const float* __restrict__ vnrm,
                                                      float* __restrict__ out) {
  int gtid = blockIdx.x * 256 + threadIdx.x;
  int wave = gtid >> 5;
  int lane = gtid & 31;
  int u = (wave % (RW / 8)) * 8 + (lane & 7);
  int v = (wave / (RW / 8)) * 4 + (lane >> 3);
  int pix = v * RW + u;

  float4 d4 = dirs[pix];
  const float dx = d4.x, dy = d4.y, dz = d4.z;
  const float ox = vm[3], oy = vm[7], oz = vm[11];  // view_matrix[:3,3]

  // initial sample at t = DEPTH_MIN
  float psdf;
  {
    float px = fmaf(DEPTH_MIN, dx, ox);
    float py = fmaf(DEPTH_MIN, dy, oy);
    float pz = fmaf(DEPTH_MIN, dz, oz);
    int ix = __float2int_rn(px), iy = __float2int_rn(py), iz = __float2int_rn(pz);
    bool inb = ((unsigned)ix < (unsigned)RD) & ((unsigned)iy < (unsigned)RD) &
               ((unsigned)iz < (unsigned)RD);
    int cx = min(max(ix, 0), RD - 1);
    int cy = min(max(iy, 0), RD - 1);
    int cz = min(max(iz, 0), RD - 1);
    uint64_t g = grid[(cx * RD + cy) * RD + cz];
    psdf = inb ? __uint_as_float((uint32_t)g) : TRUNCV;
  }

  bool hit = false;
  float depth = 0.0f;
  int hidx = -1;

  for (int i = 1; i <= NSTEPS; i += 4) {
    if (__all(hit)) break;

    float s[4];
    int gi[4];
#pragma unroll
    for (int j = 0; j < 4; ++j) {  // 4 independent loads -> MLP=4
      int st = i + j;
      float t = fmaf((float)st, INC, DEPTH_MIN);
      float px = fmaf(t, dx, ox);
      float py = fmaf(t, dy, oy);
      float pz = fmaf(t, dz, oz);
      int ix = __float2int_rn(px), iy = __float2int_rn(py), iz = __float2int_rn(pz);
      bool inb = (st <= NSTEPS) & ((unsigned)ix < (unsigned)RD) &
                 ((unsigned)iy < (unsigned)RD) & ((unsigned)iz < (unsigned)RD);
      int cx = min(max(ix, 0), RD - 1);
      int cy = min(max(iy, 0), RD - 1);
      int cz = min(max(iz, 0), RD - 1);
      uint64_t g = grid[(cx * RD + cy) * RD + cz];
      s[j]  = inb ? __uint_as_float((uint32_t)g) : TRUNCV;
      gi[j] = (int)(g >> 32);
    }

    // prefetch the first cell of the next block (gfx1250 global_prefetch_b8)
    {
      float t = fmaf((float)(i + 4), INC, DEPTH_MIN);
      int cx = min(max(__float2int_rn(fmaf(t, dx, ox)), 0), RD - 1);
      int cy = min(max(__float2int_rn(fmaf(t, dy, oy)), 0), RD - 1);
      int cz = min(max(__float2int_rn(fmaf(t, dz, oz)), 0), RD - 1);
      __builtin_prefetch((const void*)(grid + ((cx * RD + cy) * RD + cz)), 0, 0);
    }

#pragma unroll
    for (int j = 0; j < 4; ++j) {
      int st = i + j;
      if (st > NSTEPS) break;
      if (!hit) {
        float sdf = s[j];
        bool cross = (psdf > 0.0f) && (psdf < THRESH) && (sdf <= 0.0f);
        if (cross) {
          float t = fmaf((float)st, INC, DEPTH_MIN);
          float alpha = psdf / (psdf - sdf + 1e-8f);
          depth = fmaf(alpha, INC, t - INC);
          hidx = gi[j];
          hit = true;
        }
        psdf = sdf;
      }
    }
  }

  float c0 = 0, c1 = 0, c2 = 0, n0 = 0, n1 = 0, n2 = 0;
  if (hit && hidx >= 0) {
    c0 = vcol[hidx * 3 + 0]; c1 = vcol[hidx * 3 + 1]; c2 = vcol[hidx * 3 + 2];
    n0 = vnrm[hidx * 3 + 0]; n1 = vnrm[hidx * 3 + 1]; n2 = vnrm[hidx * 3 + 2];
  }
  float* out_color = out;                 // (H,W,3)
  float* out_depth = out + N_PIX * 3;     // (H,W)
  float* out_norm  = out + N_PIX * 4;     // (H,W,3)
  out_color[pix * 3 + 0] = c0;
  out_color[pix * 3 + 1] = c1;
  out_color[pix * 3 + 2] = c2;
  out_depth[pix] = depth;
  out_norm[pix * 3 + 0] = n0;
  out_norm[pix * 3 + 1] = n1;
  out_norm[pix * 3 + 2] = n2;
}

// ---------------------------------------------------------------------------
extern "C" void kernel_launch(void* const* d_in, const int* in_sizes, int n_in,
                              void* d_out, int out_size, void* d_ws, size_t ws_size,
                              hipStream_t stream) {
  const int*   locs = (const int*)d_in[0];
  const float* vsdf = (const float*)d_in[1];
  const float* vcol = (const float*)d_in[2];
  const float* vnrm = (const float*)d_in[3];
  const float* vm   = (const float*)d_in[4];
  const float* intr = (const float*)d_in[5];
  float* out = (float*)d_out;

  // Workspace layout: [0, N_CELLS*8)  packed (sdf,idx) grid
  //                   [N_CELLS*8, +N_PIX*16) float4 world-space ray directions
  uint64_t* grid = (uint64_t*)d_ws;
  float*    dirs = (float*)((char*)d_ws + (size_t)N_CELLS * 8);

  int n_locs = in_sizes[1];  // vals_sdf count == N_LOCS

  init_grid_kernel<<<(N_CELLS + 255) / 256, 256, 0, stream>>>(grid);
  scatter_kernel<<<(n_locs + 255) / 256, 256, 0, stream>>>(locs, vsdf, grid, n_locs);
  // 16 rays per wave -> N_PIX/16 waves -> *32 threads
  dir_kernel<<<((N_PIX / 16) * 32) / 256, 256, 0, stream>>>(vm, intr, dirs);
  raycast_kernel<<<N_PIX / 256, 256, 0, stream>>>(grid, (const float4*)dirs, vm, vcol,
                                                  vnrm, out);
}